// MultiHeadMemoryBank_25108378812561
// MI455X (gfx1250) — compile-verified
//
#include <hip/hip_runtime.h>
#include <math.h>

// Problem constants (match reference setup_inputs)
constexpr int B  = 64;
constexpr int N  = 16384;
constexpr int D  = 128;
constexpr int H  = 4;
constexpr int TOPK = 8;
constexpr int NKEY = 2 * H;          // 4 read + 4 write heads
constexpr int TS = 132;              // padded LDS row stride (dwords): 132 % 64 = 4 -> spreads banks

typedef __attribute__((ext_vector_type(2))) float v2f;
typedef __attribute__((ext_vector_type(8))) float v8f;

// ---------------------------------------------------------------------------
// Pass 1: sim[b, m, n] = beta[m] * <key_m, mem_n> / (||key_m||*||mem_n|| + 1e-8)
// One block = 4 waves; each wave owns a 16-row memory tile. Keys (8 real rows,
// padded to 16) form the WMMA A matrix; memory tile columns form B.
// K = 128 stepped by 4 -> 32 x v_wmma_f32_16x16x4_f32 per wave.
// Row sum-of-squares is accumulated from the same B fragments (free).
// ---------------------------------------------------------------------------
__global__ __launch_bounds__(128) void sim_norm_kernel(
    const float* __restrict__ memory, const float* __restrict__ read_keys,
    const float* __restrict__ write_keys, const float* __restrict__ beta_r,
    const float* __restrict__ beta_w, float* __restrict__ sim)
{
    __shared__ float keys[16 * TS];
    __shared__ float tile[4][16 * TS];
    __shared__ float kn[16];
    __shared__ float bt[16];

    const int b    = blockIdx.y;
    const int tid  = threadIdx.x;
    const int wave = tid >> 5;
    const int lane = tid & 31;

    // Stage 16x128 key matrix (rows 8..15 zero padding)
    for (int i = tid; i < 16 * D; i += 128) {
        const int m = i >> 7, c = i & 127;
        float v = 0.0f;
        if (m < H)           v = read_keys [((size_t)b * H + m) * D + c];
        else if (m < NKEY)   v = write_keys[((size_t)b * H + (m - H)) * D + c];
        keys[m * TS + c] = v;
    }
    __syncthreads();
    if (tid < 16) {
        float s = 0.0f;
        for (int c = 0; c < D; ++c) { const float v = keys[tid * TS + c]; s = fmaf(v, v, s); }
        kn[tid] = fmaxf(sqrtf(s), 1e-8f);
        bt[tid] = (tid < H) ? beta_r[b * H + tid]
                            : ((tid < NKEY) ? beta_w[b * H + tid - H] : 0.0f);
    }

    // Stage this wave's 16 memory rows (coalesced float4)
    const int n0 = (blockIdx.x * 4 + wave) * 16;
    const float* mrow = memory + ((size_t)b * N + n0) * D;
    for (int r = 0; r < 16; ++r) {
        const float4 v = *(const float4*)(mrow + r * D + lane * 4);
        *(float4*)&tile[wave][r * TS + lane * 4] = v;
    }
    __syncthreads();

    // WMMA fragment addressing: A 16x4 f32 and B 4x16 f32 share the pattern:
    // lane<16 holds K={kb,kb+1}, lane>=16 holds K={kb+2,kb+3}, row = lane&15.
    const int half = lane >> 4;
    const int rr   = lane & 15;
    const int koff = half * 2;
    const float* abase = &keys[rr * TS + koff];
    const float* bbase = &tile[wave][rr * TS + koff];

    v8f   acc = {};
    float ssq = 0.0f;
#pragma unroll
    for (int kb = 0; kb < D; kb += 4) {
        const v2f a  = *(const v2f*)(abase + kb);
        const v2f bb = *(const v2f*)(bbase + kb);
        acc = __builtin_amdgcn_wmma_f32_16x16x4_f32(
            /*neg_a=*/false, a, /*neg_b=*/false, bb,
            /*c_mod=*/(short)0, acc, /*reuse_a=*/false, /*reuse_b=*/false);
        ssq = fmaf(bb.x, bb.x, ssq);
        ssq = fmaf(bb.y, bb.y, ssq);
    }
    // lane l and l+16 hold complementary k-residues of the same row
    ssq += __shfl_xor(ssq, 16, 32);
    const float mn = fmaxf(sqrtf(ssq), 1e-8f);

    // D layout: lane<16, VGPR r -> (key M=r, mem row n0+lane). M=8..15 are padding.
    if (half == 0) {
        const int n = n0 + rr;
#pragma unroll
        for (int r = 0; r < NKEY; ++r) {
            const float s = bt[r] * acc[r] / (kn[r] * mn + 1e-8f);
            sim[((size_t)(b * NKEY + r)) * N + n] = s;
        }
    }
}

// ---------------------------------------------------------------------------
// Pass 2a: per (b, key-row) top-8 + softmax. 256 threads scan 16384 values.
// ---------------------------------------------------------------------------
__global__ __launch_bounds__(256) void topk_kernel(
    const float* __restrict__ sim, float* __restrict__ wsel, int* __restrict__ isel)
{
    const int row = blockIdx.x;              // b*8 + m
    const float* s = sim + (size_t)row * N;
    __shared__ float cval[2048];
    __shared__ int   cidx[2048];
    __shared__ float rval[256];
    __shared__ int   rpos[256];
    __shared__ float selv[TOPK];
    __shared__ int   seli[TOPK];

    const int t = threadIdx.x;
    float tv[TOPK]; int ti[TOPK];
#pragma unroll
    for (int j = 0; j < TOPK; ++j) { tv[j] = -INFINITY; ti[j] = 0; }
    for (int n = t; n < N; n += 256) {
        const float v = s[n];
        if (v > tv[TOPK - 1]) {
            int j = TOPK - 1;
            while (j > 0 && v > tv[j - 1]) { tv[j] = tv[j - 1]; ti[j] = ti[j - 1]; --j; }
            tv[j] = v; ti[j] = n;
        }
    }
#pragma unroll
    for (int j = 0; j < TOPK; ++j) { cval[t * TOPK + j] = tv[j]; cidx[t * TOPK + j] = ti[j]; }
    __syncthreads();

    for (int round = 0; round < TOPK; ++round) {
        float bv = -INFINITY; int bp = 0;
        for (int i = t; i < 2048; i += 256)
            if (cval[i] > bv) { bv = cval[i]; bp = i; }
        rval[t] = bv; rpos[t] = bp;
        __syncthreads();
        for (int sft = 128; sft > 0; sft >>= 1) {
            if (t < sft && rval[t + sft] > rval[t]) { rval[t] = rval[t + sft]; rpos[t] = rpos[t + sft]; }
            __syncthreads();
        }
        if (t == 0) {
            selv[round] = rval[0];
            seli[round] = cidx[rpos[0]];
            cval[rpos[0]] = -INFINITY;
        }
        __syncthreads();
    }
    if (t == 0) {
        const float m = selv[0];
        float e[TOPK], sum = 0.0f;
        for (int j = 0; j < TOPK; ++j) { e[j] = __expf(selv[j] - m); sum += e[j]; }
        const float inv = 1.0f / sum;
        for (int j = 0; j < TOPK; ++j) {
            wsel[row * TOPK + j] = e[j] * inv;
            isel[row * TOPK + j] = seli[j];
        }
    }
}

// ---------------------------------------------------------------------------
// Pass 2b: read = sum_k w * mem[idx]; read_combined = read_flat @ W_merge + b.
// One block per batch, 128 threads (one output element each).
// ---------------------------------------------------------------------------
__global__ __launch_bounds__(128) void read_merge_kernel(
    const float* __restrict__ memory, const float* __restrict__ Wm,
    const float* __restrict__ bm, const float* __restrict__ wsel,
    const int* __restrict__ isel, float* __restrict__ out)
{
    const int b = blockIdx.x;
    const int d = threadIdx.x;
    __shared__ float rf[H * D];

    for (int h = 0; h < H; ++h) {
        float acc = 0.0f;
        for (int j = 0; j < TOPK; ++j) {
            const float w = wsel[(b * NKEY + h) * TOPK + j];
            const int   n = isel[(b * NKEY + h) * TOPK + j];
            acc = fmaf(w, memory[((size_t)b * N + n) * D + d], acc);
        }
        rf[h * D + d] = acc;
    }
    __syncthreads();
    float acc = bm[d];
    for (int i = 0; i < H * D; ++i)
        acc = fmaf(rf[i], Wm[(size_t)i * D + d], acc);
    out[(size_t)b * (N + 1) * D + d] = acc;   // output row 0
}

// ---------------------------------------------------------------------------
// Pass 3: new_mem = normalize(mem*(1-erase_mean) + add_mean + 1e-8).
// Streaming: one row per wave iteration, float4 per lane, sparse entries
// (<=32 per batch) broadcast from LDS.
// ---------------------------------------------------------------------------
__global__ __launch_bounds__(256) void update_kernel(
    const float* __restrict__ memory, const float* __restrict__ write_vals,
    const float* __restrict__ erase, const float* __restrict__ add_gate,
    const float* __restrict__ wsel, const int* __restrict__ isel,
    float* __restrict__ out)
{
    const int b    = blockIdx.y;
    const int tid  = threadIdx.x;
    const int lane = tid & 31;
    const int wave = tid >> 5;

    __shared__ int   eidx[H * TOPK];
    __shared__ float ecoef[H * TOPK];
    __shared__ float acoef[H * TOPK];
    __shared__ int   ehead[H * TOPK];
    __shared__ float wv[H * D];

    for (int i = tid; i < H * D; i += 256)
        wv[i] = write_vals[(size_t)b * H * D + i];
    if (tid < H * TOPK) {
        const int h = tid >> 3, j = tid & 7;
        const int m = H + h;   // write heads occupy key rows 4..7
        const float w = wsel[(b * NKEY + m) * TOPK + j];
        eidx[tid]  = isel[(b * NKEY + m) * TOPK + j];
        ecoef[tid] = (1.0f / H) * w * erase[b * H + h];
        acoef[tid] = (1.0f / H) * w * add_gate[b * H + h];
        ehead[tid] = h;
    }
    __syncthreads();

    const int nbase = blockIdx.x * 64 + wave * 8;
    for (int r = 0; r < 8; ++r) {
        const int n = nbase + r;
        const float4 m4 = *(const float4*)(memory + ((size_t)b * N + n) * D + lane * 4);
        float es = 0.0f, ax = 0.0f, ay = 0.0f, az = 0.0f, aw = 0.0f;
#pragma unroll 1
        for (int j = 0; j < H * TOPK; ++j) {
            if (eidx[j] == n) {
                es += ecoef[j];
                const float a = acoef[j];
                const float* w4 = &wv[ehead[j] * D + lane * 4];
                ax = fmaf(a, w4[0], ax); ay = fmaf(a, w4[1], ay);
                az = fmaf(a, w4[2], az); aw = fmaf(a, w4[3], aw);
            }
        }
        const float sc = 1.0f - es;
        const float x0 = fmaf(m4.x, sc, ax) + 1e-8f;
        const float x1 = fmaf(m4.y, sc, ay) + 1e-8f;
        const float x2 = fmaf(m4.z, sc, az) + 1e-8f;
        const float x3 = fmaf(m4.w, sc, aw) + 1e-8f;
        float ssq = x0 * x0 + x1 * x1 + x2 * x2 + x3 * x3;
#pragma unroll
        for (int off = 16; off > 0; off >>= 1) ssq += __shfl_xor(ssq, off, 32);
        const float rn = 1.0f / fmaxf(sqrtf(ssq), 1e-12f);
        float4 o; o.x = x0 * rn; o.y = x1 * rn; o.z = x2 * rn; o.w = x3 * rn;
        *(float4*)(out + ((size_t)b * (N + 1) + (n + 1)) * D + lane * 4) = o;
    }
}

// ---------------------------------------------------------------------------
extern "C" void kernel_launch(void* const* d_in, const int* in_sizes, int n_in,
                              void* d_out, int out_size, void* d_ws, size_t ws_size,
                              hipStream_t stream) {
    const float* memory     = (const float*)d_in[0];
    const float* read_keys  = (const float*)d_in[1];
    const float* write_keys = (const float*)d_in[2];
    const float* write_vals = (const float*)d_in[3];
    const float* erase      = (const float*)d_in[4];
    const float* add_gate   = (const float*)d_in[5];
    const float* beta_r     = (const float*)d_in[6];
    const float* beta_w     = (const float*)d_in[7];
    const float* W_merge    = (const float*)d_in[8];
    const float* b_merge    = (const float*)d_in[9];
    float* out = (float*)d_out;

    // Workspace layout: sim[B*8*N] f32 (32 MB) | wsel[B*8*8] f32 | isel[B*8*8] i32
    float* sim  = (float*)d_ws;
    float* wsel = sim + (size_t)B * NKEY * N;
    int*   isel = (int*)(wsel + (size_t)B * NKEY * TOPK);

    (void)in_sizes; (void)n_in; (void)out_size; (void)ws_size;

    dim3 g1(N / 64, B);
    sim_norm_kernel<<<g1, 128, 0, stream>>>(memory, read_keys, write_keys,
                                            beta_r, beta_w, sim);
    topk_kernel<<<B * NKEY, 256, 0, stream>>>(sim, wsel, isel);
    read_merge_kernel<<<B, 128, 0, stream>>>(memory, W_merge, b_merge, wsel, isel, out);
    dim3 g3(N / 64, B);
    update_kernel<<<g3, 256, 0, stream>>>(memory, write_vals, erase, add_gate,
                                          wsel, isel, out);
}